// LIFConstantCurrentEncoder_26422638805214
// MI455X (gfx1250) — compile-verified
//
#include <hip/hip_runtime.h>

// LIF constant-current encoder for MI455X (gfx1250).
// Store-bandwidth-bound: 256 MB of output (voltages + spikes) vs 1 MB input.
// Strategy: 4 elems/thread in registers, 128-step loop, B128 non-temporal
// stores (CDNA5 TH=NT cache hint) since output > 192MB L2 and is never reread.

typedef float v4f __attribute__((ext_vector_type(4)));

#define SEQ_LENGTH 128
#define NELEM (512 * 512)
#define VEC 4
#define BLOCK 256  // 8 wave32 waves per workgroup

__global__ __launch_bounds__(BLOCK)
void LIFConstantCurrentEncoder_26422638805214_kernel(const float* __restrict__ x,
                                                     float* __restrict__ out) {
    const float kdt = 0.1f;  // DT * TAU_MEM_INV = 0.001 * 100
    const int tid  = blockIdx.x * BLOCK + threadIdx.x;   // 0 .. NELEM/VEC - 1
    const int base = tid * VEC;

    // One coalesced 128-bit load of the constant current per thread.
    v4f xv = *reinterpret_cast<const v4f*>(x + base);
    v4f v  = {0.0f, 0.0f, 0.0f, 0.0f};

    float* __restrict__ out_v = out;                                  // [128][512*512]
    float* __restrict__ out_z = out + (size_t)SEQ_LENGTH * NELEM;     // [128][512*512]

#pragma unroll 4
    for (int t = 0; t < SEQ_LENGTH; ++t) {
        v4f z;
#pragma unroll
        for (int k = 0; k < VEC; ++k) {
            // dv = dt*tau_mem_inv*((v_leak - v) + x);  v_leak = 0
            float dv = kdt * ((0.0f - v[k]) + xv[k]);
            float vv = v[k] + dv;
            // z = heaviside(v - v_th), v_th = 1
            float zz = (vv - 1.0f > 0.0f) ? 1.0f : 0.0f;
            // v = v - z*(v - v_reset), v_reset = 0
            vv = vv - zz * vv;
            v[k] = vv;
            z[k] = zz;
        }
        const size_t off = (size_t)t * NELEM + (size_t)base;
        // Non-temporal B128 stores: write-once data, bypass cache retention.
        __builtin_nontemporal_store(v, reinterpret_cast<v4f*>(out_v + off));
        __builtin_nontemporal_store(z, reinterpret_cast<v4f*>(out_z + off));
    }
}

extern "C" void kernel_launch(void* const* d_in, const int* in_sizes, int n_in,
                              void* d_out, int out_size, void* d_ws, size_t ws_size,
                              hipStream_t stream) {
    (void)n_in; (void)out_size; (void)d_ws; (void)ws_size;
    const float* x = (const float*)d_in[0];
    float* out = (float*)d_out;

    const int n = in_sizes[0];                 // 512*512 = 262144
    const int threads = n / VEC;               // 65536
    const int blocks  = (threads + BLOCK - 1) / BLOCK;  // 256

    LIFConstantCurrentEncoder_26422638805214_kernel<<<blocks, BLOCK, 0, stream>>>(x, out);
}